// ArcNegFace_5669356835211
// MI455X (gfx1250) — compile-verified
//
#include <hip/hip_runtime.h>
#include <math.h>

// ---- problem constants (from reference) ----
#define NB    512        // batch
#define DD    128        // feature dim (K)
#define NCLS  100000     // classes (N)

#define C_THRESH  (-0.8775825618903728f)   // cos(pi - 0.5)
#define C_COSM    ( 0.8775825618903728f)   // cos(0.5)
#define C_SINM    ( 0.4794255386042030f)   // sin(0.5)
#define C_MM      ( 0.2397127693021015f)   // sin(pi-0.5)*0.5
#define C_SCALE   64.0f
#define C_ALPHA   1.2f
#define C_INVSIG  0.5f                     // 1/SIGMA

// ---- GEMM tiling ----
#define TILE_M 64
#define TILE_N 32
#define LPAD   132       // padded LDS row stride in floats (128 + 4)
#define GBDIM  128       // GEMM block: 128 threads = 4 waves

typedef float v2f __attribute__((ext_vector_type(2)));
typedef float v8f __attribute__((ext_vector_type(8)));

__device__ __forceinline__ float wave_reduce_sum(float v) {
  v += __shfl_xor(v, 16, 32);
  v += __shfl_xor(v,  8, 32);
  v += __shfl_xor(v,  4, 32);
  v += __shfl_xor(v,  2, 32);
  v += __shfl_xor(v,  1, 32);
  return v;
}

// One wave32 per row: 128 floats -> inverse L2 norm.
__global__ void rownorm_kernel(const float* __restrict__ src,
                               float* __restrict__ dst, int rows) {
  const int row  = blockIdx.x * 8 + (threadIdx.x >> 5);
  const int lane = threadIdx.x & 31;
  if (row >= rows) return;
  const float4 f = *(const float4*)(src + (size_t)row * DD + lane * 4);
  float s = f.x * f.x + f.y * f.y + f.z * f.z + f.w * f.w;
  s = wave_reduce_sum(s);
  if (lane == 0) dst[row] = rsqrtf(s);
}

// One wave32 per batch row: label cosine -> margin-adjusted a_lb.
__global__ void alb_kernel(const float* __restrict__ feats,
                           const int*   __restrict__ labels,
                           const float* __restrict__ weight,
                           const float* __restrict__ invf,
                           const float* __restrict__ invw,
                           float* __restrict__ alb) {
  const int row  = blockIdx.x * 8 + (threadIdx.x >> 5);
  const int lane = threadIdx.x & 31;
  if (row >= NB) return;
  const int lab = labels[row];
  const float4 a = *(const float4*)(feats  + (size_t)row * DD + lane * 4);
  const float4 b = *(const float4*)(weight + (size_t)lab * DD + lane * 4);
  float s = a.x * b.x + a.y * b.y + a.z * b.z + a.w * b.w;
  s = wave_reduce_sum(s);
  if (lane == 0) {
    const float c = s * invf[row] * invw[lab];
    float av;
    if (c > C_THRESH) {
      const float cc = fminf(fmaxf(c, -1.0f), 1.0f);
      av = cc * C_COSM - sqrtf(fmaxf(1.0f - cc * cc, 0.0f)) * C_SINM;
    } else {
      av = c - C_MM;
    }
    alb[row] = av;
  }
}

// Fused GEMM (fp32 WMMA 16x16x4) + ArcNegFace epilogue.
// Block: 128 threads = 4 waves; block tile 64(M) x 32(N), K = 128 fully in LDS.
// Wave w computes the 16x32 strip at mW = w*16 using TWO independent
// accumulators (two WMMA chains -> keeps the XDL pipe full, A frag reused 2x).
__global__ __launch_bounds__(GBDIM)
void arcneg_gemm_kernel(const float* __restrict__ feats,
                        const int*   __restrict__ labels,
                        const float* __restrict__ weight,
                        const float* __restrict__ invf,
                        const float* __restrict__ invw,
                        const float* __restrict__ alb,
                        float* __restrict__ out) {
  __shared__ float lA[TILE_M * LPAD];   // feats tile, pre-scaled by invf
  __shared__ float lB[TILE_N * LPAD];   // weight tile (rows = classes)

  const int t  = threadIdx.x;
  const int n0 = blockIdx.x * TILE_N;   // class offset
  const int m0 = blockIdx.y * TILE_M;   // batch offset

  // ---- stage A: 64x128 floats, invf folded in (16 float4 / thread) ----
#pragma unroll
  for (int i = 0; i < 16; ++i) {
    const int q   = t + i * GBDIM;        // float4 index, 0..2047
    const int row = q >> 5;               // 0..63
    const int c4  = (q & 31) * 4;         // 0..124
    const float4 v = *(const float4*)(feats + (size_t)(m0 + row) * DD + c4);
    const float s  = invf[m0 + row];
    float* dst = &lA[row * LPAD + c4];
    dst[0] = v.x * s; dst[1] = v.y * s; dst[2] = v.z * s; dst[3] = v.w * s;
  }
  // ---- stage B: 32x128 floats (raw weight rows, 8 float4 / thread) ----
#pragma unroll
  for (int i = 0; i < 8; ++i) {
    const int q   = t + i * GBDIM;        // 0..1023
    const int row = q >> 5;               // 0..31
    const int c4  = (q & 31) * 4;
    const float4 v = *(const float4*)(weight + (size_t)(n0 + row) * DD + c4);
    float* dst = &lB[row * LPAD + c4];
    dst[0] = v.x; dst[1] = v.y; dst[2] = v.z; dst[3] = v.w;
  }
  __syncthreads();

  const int wave = t >> 5;
  const int lane = t & 31;
  const int mW   = wave * 16;             // wave M offset inside tile
  const int half = lane >> 4;             // 0 or 1
  const int nl   = lane & 15;             // m-index for A frag, n-index for B frag

  // fp32 WMMA fragment addressing (ISA 7.12.2):
  //   A 16x4 : a = { A[m][2h], A[m][2h+1] }     m = lane&15, h = lane>>4
  //   B 4x16 : b = { B[2h][n], B[2h+1][n] }     n = lane&15
  const float* Ab  = &lA[(mW + nl) * LPAD + 2 * half];
  const float* Bb0 = &lB[nl * LPAD + 2 * half];          // cols n0..n0+15
  const float* Bb1 = &lB[(16 + nl) * LPAD + 2 * half];   // cols n0+16..n0+31

  v8f acc0 = {};
  v8f acc1 = {};
#pragma unroll 8
  for (int kk = 0; kk < DD / 4; ++kk) {
    const int k = kk * 4;
    const v2f a  = *(const v2f*)(Ab  + k);
    const v2f b0 = *(const v2f*)(Bb0 + k);
    const v2f b1 = *(const v2f*)(Bb1 + k);
    acc0 = __builtin_amdgcn_wmma_f32_16x16x4_f32(
        false, a, false, b0, (short)0, acc0, false, false);
    acc1 = __builtin_amdgcn_wmma_f32_16x16x4_f32(
        false, a, false, b1, (short)0, acc1, false, false);
  }

  // ---- epilogue ----
  // C/D layout: acc[r] holds D[mW + r + 8*half][nT + nl]
  float albv[8];
  int   labv[8];
#pragma unroll
  for (int r = 0; r < 8; ++r) {
    const int m = m0 + mW + half * 8 + r;
    albv[r] = alb[m];
    labv[r] = labels[m];
  }
  const int   nA  = n0 + nl;
  const int   nB  = n0 + 16 + nl;
  const float iwA = invw[nA];
  const float iwB = invw[nB];

#pragma unroll
  for (int r = 0; r < 8; ++r) {
    const int   m    = m0 + mW + half * 8 + r;
    const size_t rowb = (size_t)m * NCLS;

    {
      const float c = acc0[r] * iwA;      // cosine(m, nA)
      float val;
      if (labv[r] == nA) {
        val = C_SCALE * albv[r];
      } else {
        const float d  = c - albv[r];
        const float rw = C_ALPHA * __expf(-(d * d) * C_INVSIG);
        val = C_SCALE * (rw * c + rw - 1.0f);
      }
      __builtin_nontemporal_store(val, out + rowb + nA);
    }
    {
      const float c = acc1[r] * iwB;      // cosine(m, nB)
      float val;
      if (labv[r] == nB) {
        val = C_SCALE * albv[r];
      } else {
        const float d  = c - albv[r];
        const float rw = C_ALPHA * __expf(-(d * d) * C_INVSIG);
        val = C_SCALE * (rw * c + rw - 1.0f);
      }
      __builtin_nontemporal_store(val, out + rowb + nB);
    }
  }
}

extern "C" void kernel_launch(void* const* d_in, const int* in_sizes, int n_in,
                              void* d_out, int out_size, void* d_ws, size_t ws_size,
                              hipStream_t stream) {
  const float* feats  = (const float*)d_in[0];   // [512,128]
  const int*   labels = (const int*)d_in[1];     // [512]
  const float* weight = (const float*)d_in[2];   // [100000,128]
  float* out = (float*)d_out;                    // [512,100000]

  float* wsf  = (float*)d_ws;
  float* invw = wsf;                 // 100000 floats
  float* invf = wsf + NCLS;          // 512 floats
  float* albp = wsf + NCLS + NB;     // 512 floats

  rownorm_kernel<<<(NCLS + 7) / 8, 256, 0, stream>>>(weight, invw, NCLS);
  rownorm_kernel<<<(NB + 7) / 8, 256, 0, stream>>>(feats, invf, NB);
  alb_kernel<<<(NB + 7) / 8, 256, 0, stream>>>(feats, labels, weight,
                                               invf, invw, albp);

  dim3 grid(NCLS / TILE_N, NB / TILE_M);   // (3125, 8)
  arcneg_gemm_kernel<<<grid, GBDIM, 0, stream>>>(feats, labels, weight,
                                                 invf, invw, albp, out);
}